// A2C_42734924595322
// MI455X (gfx1250) — compile-verified
//
#include <hip/hip_runtime.h>

#ifndef __has_builtin
#define __has_builtin(x) 0
#endif

#if __has_builtin(__builtin_amdgcn_global_load_async_to_lds_b128)
#define HAVE_ASYNC_LDS 1
#else
#define HAVE_ASYNC_LDS 0
#endif
#if __has_builtin(__builtin_amdgcn_s_wait_asynccnt)
#define HAVE_WAIT_ASYNC 1
#else
#define HAVE_WAIT_ASYNC 0
#endif

#define AS1 __attribute__((address_space(1)))
#define AS3 __attribute__((address_space(3)))

typedef int v4i __attribute__((vector_size(16)));  // matches builtin's int __vector(4)

constexpr int   BLOCK = 256;           // 8 wave32 waves
constexpr int   NWAVE = BLOCK / 32;    // 8
constexpr int   SEG   = 4;             // elements per thread per tile
constexpr int   TILE  = BLOCK * SEG;   // 1024 elements per tile
constexpr int   TPB   = 8;             // tiles per block
constexpr int   EPB   = TILE * TPB;    // 8192 elements per block
constexpr float GMA   = 0.99f;         // gamma
constexpr float GT    = 0.99f * 0.95f; // gamma * tau

__device__ __forceinline__ long long llmin(long long a, long long b) { return a < b ? a : b; }
__device__ __forceinline__ int imin(int a, int b) { return a < b ? a : b; }

__device__ __forceinline__ float4 ident() { return make_float4(1.f, 0.f, 1.f, 0.f); }

// Affine transform composition: (l ∘ r), l is at lower indices (applied after r).
// x = (Aret, Bret, Aadv, Badv)
__device__ __forceinline__ float4 composeXf(float4 l, float4 r) {
  float4 o;
  o.x = l.x * r.x;
  o.y = fmaf(l.x, r.y, l.y);
  o.z = l.z * r.z;
  o.w = fmaf(l.z, r.w, l.w);
  return o;
}

__device__ __forceinline__ float4 shflDown4(float4 x, int d) {
  float4 o;
  o.x = __shfl_down(x.x, d, 32);
  o.y = __shfl_down(x.y, d, 32);
  o.z = __shfl_down(x.z, d, 32);
  o.w = __shfl_down(x.w, d, 32);
  return o;
}

template <int N>
__device__ __forceinline__ void waitAsync() {
#if HAVE_WAIT_ASYNC
  __builtin_amdgcn_s_wait_asynccnt(N);
#else
  asm volatile("s_wait_asynccnt %0" ::"i"(N) : "memory");
#endif
}

// Stage one full tile (TILE elems of all 4 arrays) into LDS. Each wave issues 4
// GLOBAL_LOAD_ASYNC_TO_LDS_B128 (ASYNCcnt += 4), each lane moving 16B per array.
__device__ __forceinline__ void stageFull(
    const float* __restrict__ r, const float* __restrict__ v,
    const float* __restrict__ nv, const int* __restrict__ m,
    long long tStart, float* dR, float* dV, float* dNV, int* dM, int t) {
  const int o = t * 4;
#if HAVE_ASYNC_LDS
  __builtin_amdgcn_global_load_async_to_lds_b128((AS1 v4i*)(r + tStart + o),
                                                 (AS3 v4i*)(dR + o), 0, 0);
  __builtin_amdgcn_global_load_async_to_lds_b128((AS1 v4i*)(v + tStart + o),
                                                 (AS3 v4i*)(dV + o), 0, 0);
  __builtin_amdgcn_global_load_async_to_lds_b128((AS1 v4i*)(nv + tStart + o),
                                                 (AS3 v4i*)(dNV + o), 0, 0);
  __builtin_amdgcn_global_load_async_to_lds_b128((AS1 v4i*)(m + tStart + o),
                                                 (AS3 v4i*)(dM + o), 0, 0);
#else
  __builtin_prefetch(r + tStart + o, 0, 1);   // global_prefetch_b8 (gfx1250 path)
  *(float4*)(dR + o)  = *(const float4*)(r + tStart + o);
  *(float4*)(dV + o)  = *(const float4*)(v + tStart + o);
  *(float4*)(dNV + o) = *(const float4*)(nv + tStart + o);
  *(int4*)(dM + o)    = *(const int4*)(m + tStart + o);
#endif
}

// Per-element affine coefficients from staged tile data.
__device__ __forceinline__ void elemCoef(
    const float* sR, const float* sV, const float* sNV, const int* sM,
    int j, long long tStart, int tileLen, long long n, const float* __restrict__ gV,
    float& a_r, float& b_r, float& a_a, float& b_a) {
  const float rr  = sR[j];
  const float vv  = sV[j];
  const float nvv = sNV[j];
  const float mm  = (float)sM[j];
  float vnext;
  if (j + 1 < tileLen) {
    vnext = sV[j + 1];
  } else {
    const long long gi = tStart + j + 1;
    vnext = (gi < n) ? gV[gi] : 0.f;  // cheap scalar: values fit in L2
  }
  a_r = GMA * mm;                          // ret[i] = a_r*ret[i+1] + b_r
  b_r = fmaf(GMA * (1.f - mm), nvv, rr);
  a_a = GT * mm;                           // adv[i] = a_a*adv[i+1] + b_a
  b_a = fmaf(GMA * mm, vnext, rr) - vv;    // delta
}

struct ScanRes {
  float4 excl;   // T_{t+1} ∘ ... ∘ T_{BLOCK-1}   (identity for t = BLOCK-1)
  float4 total;  // T_0 ∘ ... ∘ T_{BLOCK-1}
};

// Block-wide suffix scan of affine transforms: wave32 shuffle butterfly (no
// barriers) + one LDS exchange of the 8 wave totals (single barrier).
__device__ __forceinline__ ScanRes blockSuffixScan(float4 my, float4* sW, int t) {
  const int lane = t & 31;
  const int wave = t >> 5;

  float4 I = my;  // becomes inclusive intra-wave suffix: T_{w,lane} ∘ ... ∘ T_{w,31}
#pragma unroll
  for (int d = 1; d < 32; d <<= 1) {
    float4 y = shflDown4(I, d);
    if (lane + d < 32) I = composeXf(I, y);
  }
  const float4 Iex = shflDown4(I, 1);  // inclusive suffix of lane+1 (valid if lane<31)

  if (lane == 0) sW[wave] = I;  // wave total
  __syncthreads();

  float4 Xw = ident();  // exclusive suffix over later waves
  for (int w2 = wave + 1; w2 < NWAVE; ++w2) Xw = composeXf(Xw, sW[w2]);
  float4 total = sW[0];
#pragma unroll
  for (int w2 = 1; w2 < NWAVE; ++w2) total = composeXf(total, sW[w2]);

  ScanRes res;
  res.excl  = composeXf((lane < 31) ? Iex : ident(), Xw);
  res.total = total;
  return res;
}

template <bool FINAL>
__device__ __forceinline__ void gae_body(
    const float* __restrict__ r, const float* __restrict__ v,
    const float* __restrict__ nv, const int* __restrict__ m,
    float4* __restrict__ blockXf, const float2* __restrict__ blockCarry,
    float* __restrict__ out, long long n) {
  __shared__ float  sR[2][TILE];
  __shared__ float  sV[2][TILE];
  __shared__ float  sNV[2][TILE];
  __shared__ int    sM[2][TILE];
  __shared__ float4 sW[NWAVE];

  const int t = threadIdx.x;
  const long long blockStart = (long long)blockIdx.x * (long long)EPB;
  if (blockStart >= n) {
    if (!FINAL && t == 0) blockXf[blockIdx.x] = ident();
    return;
  }
  const long long blockLen = llmin((long long)EPB, n - blockStart);
  const int numTiles = (int)((blockLen + TILE - 1) / TILE);

  float cret = 0.f, cadv = 0.f;   // carry entering from the right of current tile
  float4 runXf = ident();          // (partials) composed block transform
  if (FINAL) {
    const float2 c = blockCarry[blockIdx.x];
    cret = c.x; cadv = c.y;
  }

  // Prologue: stage the last (highest-index) tile of this block.
  {
    const int k = numTiles - 1;
    const long long tStart = blockStart + (long long)k * TILE;
    const int tileLen = (int)llmin((long long)TILE, n - tStart);
    if (tileLen == TILE) {
      stageFull(r, v, nv, m, tStart, sR[k & 1], sV[k & 1], sNV[k & 1], sM[k & 1], t);
    } else {
      for (int j = t; j < TILE; j += BLOCK) {  // guarded sync staging for ragged tile
        const bool ok = j < tileLen;
        const long long gi = tStart + j;
        sR[k & 1][j]  = ok ? r[gi]  : 0.f;
        sV[k & 1][j]  = ok ? v[gi]  : 0.f;
        sNV[k & 1][j] = ok ? nv[gi] : 0.f;
        sM[k & 1][j]  = ok ? m[gi]  : 0;
      }
    }
  }

  for (int k = numTiles - 1; k >= 0; --k) {
    const int b = k & 1;
    if (k > 0) {  // prefetch next (lower) tile into the other buffer
      const long long pStart = blockStart + (long long)(k - 1) * TILE;
      stageFull(r, v, nv, m, pStart, sR[1 - b], sV[1 - b], sNV[1 - b], sM[1 - b], t);
    }
#if HAVE_ASYNC_LDS
    if (k > 0) waitAsync<4>(); else waitAsync<0>();  // wait only for current tile's 4 issues
#endif
    __syncthreads();

    const long long tStart = blockStart + (long long)k * TILE;
    const int tileLen = (int)llmin((long long)TILE, n - tStart);
    const int lo = t * SEG;
    const int hi = imin(lo + SEG, tileLen);

    // Serial composition of this thread's SEG elements (high -> low index).
    float4 xf = ident();
    for (int j = hi - 1; j >= lo; --j) {
      float a_r, b_r, a_a, b_a;
      elemCoef(sR[b], sV[b], sNV[b], sM[b], j, tStart, tileLen, n, v, a_r, b_r, a_a, b_a);
      xf = composeXf(make_float4(a_r, b_r, a_a, b_a), xf);
    }

    const ScanRes sc = blockSuffixScan(xf, sW, t);

    if (FINAL) {
      float tr = fmaf(sc.excl.x, cret, sc.excl.y);  // carry at right edge of this segment
      float ta = fmaf(sc.excl.z, cadv, sc.excl.w);
      float oadv[SEG], oret[SEG];
      for (int j = hi - 1; j >= lo; --j) {
        float a_r, b_r, a_a, b_a;
        elemCoef(sR[b], sV[b], sNV[b], sM[b], j, tStart, tileLen, n, v, a_r, b_r, a_a, b_a);
        tr = fmaf(a_r, tr, b_r);
        ta = fmaf(a_a, ta, b_a);
        oret[j - lo] = tr;
        oadv[j - lo] = ta;
      }
      const long long gb = tStart + lo;
      if (hi - lo == SEG) {
        *(float4*)(out + gb) = make_float4(oadv[0], oadv[1], oadv[2], oadv[3]);
        if ((n & 3LL) == 0LL) {
          *(float4*)(out + n + gb) = make_float4(oret[0], oret[1], oret[2], oret[3]);
        } else {
          for (int q = 0; q < SEG; ++q) out[n + gb + q] = oret[q];
        }
      } else {
        for (int j = lo; j < hi; ++j) {
          out[tStart + j]     = oadv[j - lo];
          out[n + tStart + j] = oret[j - lo];
        }
      }
      cret = fmaf(sc.total.x, cret, sc.total.y);  // carry for the next (lower) tile
      cadv = fmaf(sc.total.z, cadv, sc.total.w);
    } else {
      runXf = composeXf(sc.total, runXf);
    }
    __syncthreads();  // protect LDS tile buffers + sW before reuse
  }

  if (!FINAL && t == 0) blockXf[blockIdx.x] = runXf;
}

// Defined first so the async-LDS instructions show up at the top of the disasm.
__global__ __launch_bounds__(BLOCK) void gae_partial(
    const float* __restrict__ r, const float* __restrict__ v,
    const float* __restrict__ nv, const int* __restrict__ m,
    float4* __restrict__ blockXf, long long n) {
  gae_body<false>(r, v, nv, m, blockXf, nullptr, nullptr, n);
}

// Reverse-exclusive scan over block transforms -> per-block incoming carries.
__global__ __launch_bounds__(BLOCK) void gae_spine(
    const float4* __restrict__ xf, float2* __restrict__ carry, int nblk) {
  __shared__ float4 sW[NWAVE];
  const int t = threadIdx.x;
  const int pb = (nblk + BLOCK - 1) / BLOCK;
  const int lo = t * pb;
  const int hi = imin(lo + pb, nblk);

  float4 my = ident();
  for (int j = hi - 1; j >= lo; --j) my = composeXf(xf[j], my);

  const ScanRes sc = blockSuffixScan(my, sW, t);
  float cr = sc.excl.y;  // exclusive suffix applied to (0,0)
  float ca = sc.excl.w;
  for (int j = hi - 1; j >= lo; --j) {
    carry[j] = make_float2(cr, ca);
    const float4 x = xf[j];
    cr = fmaf(x.x, cr, x.y);
    ca = fmaf(x.z, ca, x.w);
  }
}

__global__ __launch_bounds__(BLOCK) void gae_final(
    const float* __restrict__ r, const float* __restrict__ v,
    const float* __restrict__ nv, const int* __restrict__ m,
    const float2* __restrict__ blockCarry, float* __restrict__ out, long long n) {
  gae_body<true>(r, v, nv, m, nullptr, blockCarry, out, n);
}

extern "C" void kernel_launch(void* const* d_in, const int* in_sizes, int n_in,
                              void* d_out, int out_size, void* d_ws, size_t ws_size,
                              hipStream_t stream) {
  const float* r  = (const float*)d_in[0];
  const float* v  = (const float*)d_in[1];
  const float* nv = (const float*)d_in[2];
  const int*   m  = (const int*)d_in[3];
  float* out = (float*)d_out;  // [advantages (n) | returns (n)]
  const long long n = (long long)in_sizes[0];
  if (n <= 0) return;

  const int nblk = (int)((n + EPB - 1) / EPB);  // 512 at T = 4M
  float4* blockXf    = (float4*)d_ws;
  float2* blockCarry = (float2*)((char*)d_ws + (size_t)nblk * sizeof(float4));

  gae_partial<<<nblk, BLOCK, 0, stream>>>(r, v, nv, m, blockXf, n);
  gae_spine<<<1, BLOCK, 0, stream>>>(blockXf, blockCarry, nblk);
  gae_final<<<nblk, BLOCK, 0, stream>>>(r, v, nv, m, blockCarry, out, n);
}